// HighOrderNet_63969242907063
// MI455X (gfx1250) — compile-verified
//
#include <hip/hip_runtime.h>

typedef __attribute__((ext_vector_type(2))) float v2f;
typedef __attribute__((ext_vector_type(8))) float v8f;

#define NROWS 1024
#define HDIM  512
#define ODIM  512
#define AS    516   // padded LDS row stride in floats (516 = 8*64+4 -> conflict-free b64 reads)

__global__ __launch_bounds__(256)
void highorder_wmma_kernel(const int*   __restrict__ x,      // (64,3)
                           const int*   __restrict__ fact,   // (1024,2)
                           const float* __restrict__ inp,    // (3,1024,512)
                           const float* __restrict__ params, // (1000,512,512)
                           const float* __restrict__ bias,   // (1000,1,512)
                           const int*   __restrict__ msg_to_p,
                           float*       __restrict__ out)    // (1024,512)
{
    __shared__ float sA[16 * AS];
    __shared__ int   sRows[NROWS];
    __shared__ int   sCnt;

    const int node  = blockIdx.y;   // 0..63
    const int otile = blockIdx.x;   // 0..3, 128 cols each
    const int tid   = threadIdx.x;
    const int lane  = tid & 31;
    const int wave  = tid >> 5;     // 0..7
    const int hi    = lane >> 4;    // half-wave id
    const int lo    = lane & 15;

    if (tid == 0) sCnt = 0;
    __syncthreads();

    // ---- gather rows mapped to this node (order-independent result) ----
    for (int n = tid; n < NROWS; n += 256) {
        if (fact[2 * n] == node) {
            int slot = atomicAdd(&sCnt, 1);
            sRows[slot] = n;
        }
    }
    __syncthreads();

    const int cnt = sCnt;
    if (cnt == 0) return;                       // uniform exit
    const int cntPad = (cnt + 15) & ~15;
    for (int i = cnt + tid; i < cntPad; i += 256) sRows[i] = -1;
    __syncthreads();

    // ---- which two tensors of inp to multiply (keep = indices != msg_to) ----
    const int mt = msg_to_p[0];                 // 0 in reference setup
    const int k1 = (mt == 0) ? 1 : 0;
    const int k2 = (mt == 2) ? 1 : 2;
    const float* inpA = inp + (size_t)k1 * NROWS * HDIM;
    const float* inpB = inp + (size_t)k2 * NROWS * HDIM;

    const int id = x[3 * node + 2];
    const float* W  = params + (size_t)id * HDIM * ODIM;
    const float* Bv = bias   + (size_t)id * ODIM;

    const int col  = otile * 128 + wave * 16 + lo;
    const float bval = Bv[col];

    // per-lane B-matrix (4x16, K-major per ISA layout) base offsets into W
    const int bx0 = (2 * hi) * ODIM + col;      // K row 2*hi
    const int by0 = (2 * hi + 1) * ODIM + col;  // K row 2*hi+1

    for (int m0 = 0; m0 < cntPad; m0 += 16) {
        // ---- stage A tile: fact_prod for 16 rows into LDS (coalesced) ----
        for (int i = tid; i < 16 * HDIM; i += 256) {
            int r = i >> 9;            // / 512
            int k = i & (HDIM - 1);
            int row = sRows[m0 + r];
            float v = 0.0f;
            if (row >= 0)
                v = inpA[(size_t)row * HDIM + k] * inpB[(size_t)row * HDIM + k];
            sA[r * AS + k] = v;
        }
        __syncthreads();

        // ---- K loop: D = A(16x4) * B(4x16) + D via V_WMMA_F32_16X16X4_F32 ----
        v8f acc = {0.f, 0.f, 0.f, 0.f, 0.f, 0.f, 0.f, 0.f};
        const float* aRow = &sA[lo * AS + 2 * hi];   // A: lanes 0-15 K={0,1}, 16-31 K={2,3}
        #pragma unroll 8
        for (int kk = 0; kk < HDIM; kk += 4) {
            v2f a = *(const v2f*)(aRow + kk);        // 8B-aligned ds_load_b64
            v2f b;
            b.x = W[bx0 + kk * ODIM];
            b.y = W[by0 + kk * ODIM];
            acc = __builtin_amdgcn_wmma_f32_16x16x4_f32(
                /*neg_a=*/false, a, /*neg_b=*/false, b,
                /*c_mod=*/(short)0, acc, /*reuse_a=*/false, /*reuse_b=*/false);
        }

        // ---- writeback: C/D vgpr r -> M = r + 8*hi, N = col ----
        #pragma unroll
        for (int r = 0; r < 8; ++r) {
            int row = sRows[m0 + r + 8 * hi];
            if (row >= 0)
                out[(size_t)row * ODIM + col] = acc[r] + bval;
        }
        __syncthreads();
    }
}

extern "C" void kernel_launch(void* const* d_in, const int* in_sizes, int n_in,
                              void* d_out, int out_size, void* d_ws, size_t ws_size,
                              hipStream_t stream) {
    const int*   x      = (const int*)d_in[0];
    const int*   fact   = (const int*)d_in[1];
    const float* inp    = (const float*)d_in[2];
    const float* params = (const float*)d_in[3];
    const float* bias   = (const float*)d_in[4];
    const int*   msg_to = (const int*)d_in[5];
    float*       out    = (float*)d_out;

    dim3 grid(4, 64);   // 4 o-tiles of 128 cols x 64 nodes
    highorder_wmma_kernel<<<grid, 256, 0, stream>>>(x, fact, inp, params, bias, msg_to, out);
}